// Caps_Layer_738734375579
// MI455X (gfx1250) — compile-verified
//
#include <hip/hip_runtime.h>
#include <hip/hip_bf16.h>

#if defined(__HIP_DEVICE_COMPILE__) && !__has_builtin(__builtin_amdgcn_wmma_f32_16x16x4_f32)
#error "missing __builtin_amdgcn_wmma_f32_16x16x4_f32 on device toolchain"
#endif

typedef float v2f __attribute__((ext_vector_type(2)));
typedef float v8f __attribute__((ext_vector_type(8)));

#define BATCH 512
#define TLEN  2048
#define DIN   100
#define NCAP  5
#define DCAP  5
#define NOUT  25          // NCAP*DCAP
#define ROUTINGS 4
#define CAP_EPS 1e-7f

#define UH_STRIDE 26      // even (8B-aligned rows) + conflict-free for b64 column reads
#define THREADS 256
#define WAVES   8
#define KCHUNKS (DIN / 4) // 25

// dynamic LDS partition (floats)
#define OFF_UHAT   0                                   // TLEN*26          = 53248
#define OFF_W      (OFF_UHAT + TLEN * UH_STRIDE)       // 25*2*32 float2   = 3200 floats (even offset -> 8B aligned)
#define OFF_RED    (OFF_W + KCHUNKS * 2 * 32 * 2)      // 8*25             = 200
#define OFF_OUTS   (OFF_RED + WAVES * NOUT)            // 32
#define OFF_TRASH  (OFF_OUTS + 32)                     // 32 (dump for invalid D-store lanes)
#define SMEM_FLOATS (OFF_TRASH + 32)
#define SMEM_BYTES  (SMEM_FLOATS * 4)

__global__ __launch_bounds__(THREADS) void caps_layer_kernel(
    const float* __restrict__ x,   // (B, T, D)
    const float* __restrict__ W,   // (1, D, 25)
    float* __restrict__ out)       // (B, 5, 5)
{
    extern __shared__ float smem[];
    float* u_lds  = smem + OFF_UHAT;   // [t][26], cols 0..24 = u_hat[b,t,n*5+d]
    v2f*   Wf2    = (v2f*)(smem + OFF_W); // fragment-interleaved W: [(kk*2+half)*32 + col] -> (K=4kk+2half, K+1)
    float* red    = smem + OFF_RED;    // [wave][25]
    float* outs   = smem + OFF_OUTS;   // [25] squashed outputs (double-use across iters)

    const int tid  = threadIdx.x;
    const int bIdx = blockIdx.x;
    const int lane = tid & 31;
    const int wave = tid >> 5;
    const int half = lane >> 4;   // 0: lanes 0-15, 1: lanes 16-31
    const int lrow = lane & 15;

    // ---- stage W into LDS, fragment-interleaved + zero-padded to 32 cols ----
    // element i: col = i&31, hc = (i>>5)&1, kk = i>>6 ; holds W rows (4kk+2hc, 4kk+2hc+1)
    for (int i = tid; i < KCHUNKS * 2 * 32; i += THREADS) {
        const int col = i & 31, hc = (i >> 5) & 1, kk = i >> 6;
        const int k0 = 4 * kk + 2 * hc;
        v2f v; v.x = 0.0f; v.y = 0.0f;
        if (col < NOUT) { v.x = W[k0 * NOUT + col]; v.y = W[(k0 + 1) * NOUT + col]; }
        Wf2[i] = v;
    }
    __syncthreads();

    // ---- GEMM: u_hat[b] = x[b] @ W via V_WMMA_F32_16X16X4_F32 ----
    // A (16x4 f32): v0 = A[M=lrow, K=2*half], v1 = K=2*half+1  -> one float2 global load
    // B (4x16 f32): v0 = B[K=2*half, N],  v1 = B[K=2*half+1, N] -> one ds_load_b64
    // D (16x16 f32): vgpr i -> row = t0 + i + 8*half, col = N
    const float* xb = x + (size_t)bIdx * TLEN * DIN;
    for (int tile = wave; tile < TLEN / 16; tile += WAVES) {
        const int t0 = tile * 16;
        const float* arow = xb + (size_t)(t0 + lrow) * DIN + 2 * half;
        // prefetch this wave's next tile rows
        __builtin_prefetch(arow + (size_t)(16 * WAVES) * DIN, 0, 1);
        __builtin_prefetch(arow + (size_t)(16 * WAVES) * DIN + 64, 0, 1);
        v8f d0 = {}; v8f d1 = {};
#pragma unroll
        for (int kk = 0; kk < KCHUNKS; ++kk) {
            v2f a  = *(const v2f*)(arow + 4 * kk);
            v2f b0 = Wf2[(kk * 2 + half) * 32 + lrow];
            v2f b1 = Wf2[(kk * 2 + half) * 32 + 16 + lrow];
#if defined(__HIP_DEVICE_COMPILE__)
            d0 = __builtin_amdgcn_wmma_f32_16x16x4_f32(false, a, false, b0, (short)0, d0, false, false);
            d1 = __builtin_amdgcn_wmma_f32_16x16x4_f32(false, a, false, b1, (short)0, d1, false, false);
#else
            (void)a; (void)b0; (void)b1;
#endif
        }
        const int rbase = t0 + 8 * half;
        const int c1 = 16 + lrow;                 // cols 16..31; >=25 invalid
#pragma unroll
        for (int i = 0; i < 8; ++i) {
            u_lds[(rbase + i) * UH_STRIDE + lrow] = d0[i];
            const int a1 = (c1 < NOUT) ? ((rbase + i) * UH_STRIDE + c1) : (OFF_TRASH - OFF_UHAT);
            u_lds[a1] = d1[i];                    // branchless: invalid lanes hit trash slot
        }
    }
    __syncthreads();

    // ---- dynamic routing, fused (no stored b): bn recomputed from prev outputs ----
    for (int iter = 0; iter < ROUTINGS; ++iter) {
        float on[NOUT];
        if (iter > 0) {
#pragma unroll
            for (int j = 0; j < NOUT; ++j) on[j] = outs[j];  // LDS broadcast reads
        }

        float acc[NOUT];
#pragma unroll
        for (int j = 0; j < NOUT; ++j) acc[j] = 0.0f;

        for (int t = tid; t < TLEN; t += THREADS) {
            float u[26];
            const v2f* ur2 = (const v2f*)(u_lds + t * UH_STRIDE);
#pragma unroll
            for (int m = 0; m < 13; ++m) { v2f p = ur2[m]; u[2 * m] = p.x; u[2 * m + 1] = p.y; }

            float cn[NCAP];
            if (iter == 0) {
#pragma unroll
                for (int n = 0; n < NCAP; ++n) cn[n] = 1.0f / NCAP;
            } else {
                // b[n,t] = sum_d outs_prev[n,d] * u_hat[n,t,d], then softmax over n
                float bn[NCAP], m = -3.4e38f;
#pragma unroll
                for (int n = 0; n < NCAP; ++n) {
                    float s = 0.0f;
#pragma unroll
                    for (int d = 0; d < DCAP; ++d) s += on[n * DCAP + d] * u[n * DCAP + d];
                    bn[n] = s; m = fmaxf(m, s);
                }
                float ssum = 0.0f;
#pragma unroll
                for (int n = 0; n < NCAP; ++n) { cn[n] = __expf(bn[n] - m); ssum += cn[n]; }
                float inv = 1.0f / ssum;
#pragma unroll
                for (int n = 0; n < NCAP; ++n) cn[n] *= inv;
            }
#pragma unroll
            for (int n = 0; n < NCAP; ++n)
#pragma unroll
                for (int d = 0; d < DCAP; ++d)
                    acc[n * DCAP + d] += cn[n] * u[n * DCAP + d];
        }

        // wave-level tree reduction (fixed order -> deterministic)
#pragma unroll
        for (int off = 16; off > 0; off >>= 1)
#pragma unroll
            for (int j = 0; j < NOUT; ++j)
                acc[j] += __shfl_down(acc[j], off, 32);
        if (lane == 0)
#pragma unroll
            for (int j = 0; j < NOUT; ++j) red[wave * NOUT + j] = acc[j];
        __syncthreads();

        // cross-wave reduce + squash, all inside wave 0 (shfl gathers the norm group)
        if (wave == 0) {
            float s = 0.0f;
            if (lane < NOUT) {
#pragma unroll
                for (int w = 0; w < WAVES; ++w) s += red[w * NOUT + lane];
            }
            const int nb = (lane / DCAP) * DCAP;
            float nrm = 0.0f;
#pragma unroll
            for (int d = 0; d < DCAP; ++d) { float v = __shfl(s, nb + d, 32); nrm += v * v; }
            if (lane < NOUT) outs[lane] = s / sqrtf(nrm + CAP_EPS);
        }
        __syncthreads();
    }

    if (tid < NOUT) out[(size_t)bIdx * NOUT + tid] = outs[tid];
}

extern "C" void kernel_launch(void* const* d_in, const int* in_sizes, int n_in,
                              void* d_out, int out_size, void* d_ws, size_t ws_size,
                              hipStream_t stream) {
    const float* x = (const float*)d_in[0];
    const float* W = (const float*)d_in[1];
    float* out = (float*)d_out;
    (void)in_sizes; (void)n_in; (void)out_size; (void)d_ws; (void)ws_size;

    // opt in to >64KB dynamic LDS (CDNA5 WGP: up to 320KB per workgroup)
    (void)hipFuncSetAttribute((const void*)caps_layer_kernel,
                              hipFuncAttributeMaxDynamicSharedMemorySize, SMEM_BYTES);

    caps_layer_kernel<<<BATCH, THREADS, SMEM_BYTES, stream>>>(x, W, out);
}